// MaxCutLogSumExp_3702261809399
// MI455X (gfx1250) — compile-verified
//
#include <hip/hip_runtime.h>
#include <math.h>

#define WAVE            32
#define WAVES_PER_BLOCK 8
#define BLOCK           (WAVE * WAVES_PER_BLOCK)
#define MAX_M           128          // per-wave LDS staging (floats); covers M<=100
#define REGS            (MAX_M / WAVE)
#define BATCHN          256

#ifndef __has_builtin
#define __has_builtin(x) 0
#endif

#if __has_builtin(__builtin_amdgcn_global_load_async_to_lds_b32) && \
    __has_builtin(__builtin_amdgcn_global_store_async_from_lds_b32)
#define USE_ASYNC_LDS 1
typedef __attribute__((address_space(1))) int gint_t;   // global (AS1)
typedef __attribute__((address_space(3))) int lint_t;   // LDS (AS3)
#else
#define USE_ASYNC_LDS 0
#endif

#if __has_builtin(__builtin_amdgcn_s_wait_asynccnt)
#define WAIT_ASYNC_0() __builtin_amdgcn_s_wait_asynccnt(0)
#else
#define WAIT_ASYNC_0() asm volatile("s_wait_asynccnt 0" ::: "memory")
#endif

__device__ __forceinline__ int lower_bound(const int* __restrict__ a, int n, int key) {
    int lo = 0, hi = n;
    while (lo < hi) {
        int mid = (lo + hi) >> 1;
        if (a[mid] < key) lo = mid + 1; else hi = mid;
    }
    return lo;
}

__device__ __forceinline__ float wave_max(float v) {
    for (int m = 16; m >= 1; m >>= 1) v = fmaxf(v, __shfl_xor(v, m, WAVE));
    return v;
}
__device__ __forceinline__ float wave_sum(float v) {
    for (int m = 16; m >= 1; m >>= 1) v += __shfl_xor(v, m, WAVE);
    return v;
}

// Grid: x = parent slots (8 wave32 per block, consecutive waves -> consecutive
// parents -> contiguous child segments), y = batch row. No integer division.
// Children of parent i are the contiguous node range starting at
// concat_children[k_start]; row boundaries found by binary search on the
// strictly-increasing flat_idx (= i*M + j).
// If copy_children: also move child values scores->out (fused leaf copy)
// through LDS using the CDNA5 async global<->LDS DMA path (ASYNCcnt).
__global__ __launch_bounds__(BLOCK)
void level_reduce_kernel(const float* __restrict__ scores,
                         const float* __restrict__ child_src,
                         float* __restrict__ out,
                         const int* __restrict__ parents,
                         const int* __restrict__ flat_idx,
                         const int* __restrict__ concat_children,
                         const int* __restrict__ Mptr,
                         int P, int Nc, int num_nodes, int copy_children)
{
    __shared__ float lbuf[WAVES_PER_BLOCK * MAX_M];

    const int lane = threadIdx.x & (WAVE - 1);
    const int wave = threadIdx.x >> 5;
    const int i    = blockIdx.x * WAVES_PER_BLOCK + wave;  // parent slot
    if (i >= P) return;
    const int b    = blockIdx.y;                           // batch row
    const int M    = Mptr[0];                              // device-resident python scalar

    const int k_start = lower_bound(flat_idx, Nc, i * M);
    const int k_end   = lower_bound(flat_idx, Nc, (i + 1) * M);
    const int cnt     = k_end - k_start;

    const int  parent_node = parents[i];
    const long long rowbase = (long long)b * num_nodes;
    const float oldv = scores[rowbase + parent_node];

    if (cnt <= 0) {
        if (lane == 0) out[rowbase + parent_node] = oldv;
        return;
    }

    const int child0 = concat_children[k_start];      // children contiguous (BFS numbering)
    const float* __restrict__ src = child_src + rowbase + child0;
    float* __restrict__ wbuf = &lbuf[wave * MAX_M];

    float vals[REGS];
    float m = -INFINITY;

#if USE_ASYNC_LDS
    // Stage the contiguous child segment into LDS via async DMA (ASYNCcnt).
    for (int t = lane; t < cnt; t += WAVE) {
        __builtin_amdgcn_global_load_async_to_lds_b32(
            (gint_t*)(src + t), (lint_t*)(wbuf + t), 0, 0);
    }
    WAIT_ASYNC_0();
    if (copy_children) {
        float* dst = out + rowbase + child0;
        for (int t = lane; t < cnt; t += WAVE) {
            __builtin_amdgcn_global_store_async_from_lds_b32(
                (gint_t*)(dst + t), (lint_t*)(wbuf + t), 0, 0);
        }
    }
    #pragma unroll
    for (int r = 0; r < REGS; ++r) {
        const int idx = lane + WAVE * r;
        vals[r] = (idx < cnt) ? wbuf[idx] : -INFINITY;
        m = fmaxf(m, vals[r]);
    }
#else
    #pragma unroll
    for (int r = 0; r < REGS; ++r) {
        const int idx = lane + WAVE * r;
        vals[r] = (idx < cnt) ? src[idx] : -INFINITY;
        m = fmaxf(m, vals[r]);
    }
    if (copy_children) {
        float* dst = out + rowbase + child0;
        #pragma unroll
        for (int r = 0; r < REGS; ++r) {
            const int idx = lane + WAVE * r;
            if (idx < cnt) dst[idx] = vals[r];
        }
    }
#endif

    m = wave_max(m);
    float s = 0.0f;
    #pragma unroll
    for (int r = 0; r < REGS; ++r) s += expf(vals[r] - m);   // exp(-inf - m) = 0 pads
    s = wave_sum(s);

    if (lane == 0) {
        const float lse = m + logf(s);
        out[rowbase + parent_node] = fmaxf(oldv, lse);
    }
}

extern "C" void kernel_launch(void* const* d_in, const int* in_sizes, int n_in,
                              void* d_out, int out_size, void* d_ws, size_t ws_size,
                              hipStream_t stream) {
    (void)n_in; (void)out_size; (void)d_ws; (void)ws_size;

    const float* scores = (const float*)d_in[0];
    // setup_inputs order: scores, then per level d: p{d}, f{d}, c{d}, P{d}, M{d}
    const int* p[3] = {(const int*)d_in[1],  (const int*)d_in[6],  (const int*)d_in[11]};
    const int* f[3] = {(const int*)d_in[2],  (const int*)d_in[7],  (const int*)d_in[12]};
    const int* c[3] = {(const int*)d_in[3],  (const int*)d_in[8],  (const int*)d_in[13]};
    const int* M[3] = {(const int*)d_in[5],  (const int*)d_in[10], (const int*)d_in[15]};
    const int  P[3]  = {in_sizes[1], in_sizes[6],  in_sizes[11]};
    const int  Nc[3] = {in_sizes[2], in_sizes[7],  in_sizes[12]};
    const int  num_nodes = in_sizes[0] / BATCHN;

    float* out = (float*)d_out;

    // Bottom-up: d=2 (reads scores, fused leaf copy), then d=1, d=0 (read out).
    for (int d = 2; d >= 0; --d) {
        const float* child_src = (d == 2) ? scores : out;
        const int copy = (d == 2) ? 1 : 0;
        const int blocks = (P[d] + WAVES_PER_BLOCK - 1) / WAVES_PER_BLOCK;
        level_reduce_kernel<<<dim3(blocks, BATCHN), dim3(BLOCK), 0, stream>>>(
            scores, child_src, out, p[d], f[d], c[d], M[d],
            P[d], Nc[d], num_nodes, copy);
    }
}